// Proto_77687368450665
// MI455X (gfx1250) — compile-verified
//
#include <hip/hip_runtime.h>

typedef __attribute__((ext_vector_type(2))) float v2f;
typedef __attribute__((ext_vector_type(8))) float v8f;

// Problem constants (from reference)
constexpr int NS  = 25;            // WAY*SHOT       (support count, logits axis 1)
constexpr int NQ  = 75;            // WAY*QUERY_SHOT (query count,  logits axis 0)
constexpr int NQP = 80;            // padded q-stride for centroid workspace
constexpr int D   = 640;
constexpr int M   = 25;
constexpr int DT  = D / 16;        // 40 d-tiles
constexpr float INV_M       = 1.0f / 25.0f;   // centroid & query_r are /M
constexpr float LOGIT_SCALE = 1.0f / 500.0f;  // 1/(G*Q*Q*SHOT)

__device__ __forceinline__ int iclamp(int x, int hi) { return x < hi ? x : hi; }

// ---------------------------------------------------------------------------
// Phase 0: zero the l2^T accumulator (75 x 25 floats) in workspace
// ---------------------------------------------------------------------------
__global__ void zero_l2_kernel(float* __restrict__ l2t) {
    int i = blockIdx.x * blockDim.x + threadIdx.x;
    if (i < NQ * NS) l2t[i] = 0.0f;
}

// ---------------------------------------------------------------------------
// Phase 1: centroid[s,q,d] = sum_m weight2[s,q,m] * support[s,d,m]
// One wave per (s, q-tile, d-tile). GEMM tile 16(q) x 16(d), K=25 pad 28.
// All loads unconditional (clamped addresses); padding zeroed via cndmask on A
// only — B padding columns are multiplied by zeroed A columns (0*finite == 0).
// grid = (40, 5, 25), block = 32
// ---------------------------------------------------------------------------
__global__ void centroid_kernel(const float* __restrict__ fm,
                                const float* __restrict__ w2,
                                float* __restrict__ cen) {
    const int lane = threadIdx.x;
    const int dt = blockIdx.x, qt = blockIdx.y, s = blockIdx.z;
    const int r = lane & 15;      // A row (q) / B col (d)
    const int g = lane >> 4;      // half-wave: k sub-offset

    const int q  = qt * 16 + r;
    const bool qv = (q < NQ);
    const int qc = iclamp(q, NQ - 1);
    const int d  = dt * 16 + r;

    const float* wrow = w2 + (s * NQ + qc) * M;  // weight2[s,q,:] (clamped q)
    const float* srow = fm + (s * D + d) * M;    // support[s,d,:]

    v8f acc = {0.f, 0.f, 0.f, 0.f, 0.f, 0.f, 0.f, 0.f};
#pragma unroll
    for (int kk = 0; kk < 7; ++kk) {             // K = 28 (m padded 25->28)
        const int m0 = kk * 4 + g * 2;
        const int m1 = m0 + 1;
        const float a0 = wrow[iclamp(m0, M - 1)];
        const float a1 = wrow[iclamp(m1, M - 1)];
        v2f a, b;
        a.x = (qv && m0 < M) ? a0 : 0.0f;        // v_cndmask, no branch
        a.y = (qv && m1 < M) ? a1 : 0.0f;
        b.x = srow[iclamp(m0, M - 1)];           // unmasked: paired A col is 0
        b.y = srow[iclamp(m1, M - 1)];
        acc = __builtin_amdgcn_wmma_f32_16x16x4_f32(
            /*neg_a=*/false, a, /*neg_b=*/false, b,
            /*c_mod=*/(short)0, acc, /*reuse_a=*/false, /*reuse_b=*/false);
    }
    // Unconditional stores into q-padded workspace (rows >= NQ hold exact 0s).
#pragma unroll
    for (int v = 0; v < 8; ++v) {
        const int qo = qt * 16 + v + g * 8;      // C/D layout: row = v + g*8
        cen[(s * NQP + qo) * D + dt * 16 + r] = acc[v];
    }
}

// ---------------------------------------------------------------------------
// Phase 2: query_r[s,q,d] = sum_m weight2[s,q,m] * query[q,d,m] (tile in regs),
// fused with diff^2 and d-reduction into l2^T[q,s] via atomics.
// One wave per (q, s-tile, d-tile). grid = (40, 2, 75), block = 32
// ---------------------------------------------------------------------------
__global__ void qr_l2_kernel(const float* __restrict__ fm,
                             const float* __restrict__ w2,
                             const float* __restrict__ cen,
                             float* __restrict__ l2t) {
    const int lane = threadIdx.x;
    const int dt = blockIdx.x, st = blockIdx.y, q = blockIdx.z;
    const int r = lane & 15;
    const int g = lane >> 4;

    const int s  = st * 16 + r;
    const bool sv = (s < NS);
    const int sc = iclamp(s, NS - 1);
    const int d  = dt * 16 + r;

    const float* wrow = w2 + (sc * NQ + q) * M;        // weight2[s,q,:] (clamped s)
    const float* qrow = fm + ((NS + q) * D + d) * M;   // query[q,d,:]

    v8f acc = {0.f, 0.f, 0.f, 0.f, 0.f, 0.f, 0.f, 0.f};
#pragma unroll
    for (int kk = 0; kk < 7; ++kk) {
        const int m0 = kk * 4 + g * 2;
        const int m1 = m0 + 1;
        const float a0 = wrow[iclamp(m0, M - 1)];
        const float a1 = wrow[iclamp(m1, M - 1)];
        v2f a, b;
        a.x = (sv && m0 < M) ? a0 : 0.0f;
        a.y = (sv && m1 < M) ? a1 : 0.0f;
        b.x = qrow[iclamp(m0, M - 1)];
        b.y = qrow[iclamp(m1, M - 1)];
        acc = __builtin_amdgcn_wmma_f32_16x16x4_f32(
            false, a, false, b, (short)0, acc, false, false);
    }

    // diff = (centroid_raw - queryr_raw)/M ; accumulate sum_d diff^2 -> l2t[q,s]
#pragma unroll
    for (int v = 0; v < 8; ++v) {
        const int so  = st * 16 + v + g * 8;
        const bool ov = (so < NS);
        const int soc = iclamp(so, NS - 1);
        const float c = cen[(soc * NQP + q) * D + dt * 16 + r]; // unconditional
        const float diff = (c - acc[v]) * INV_M;
        float p = ov ? diff * diff : 0.0f;                      // v_cndmask
        // reduce across the 16 lanes of this half-wave (masks < 16 stay in half)
        p += __shfl_xor(p, 1);
        p += __shfl_xor(p, 2);
        p += __shfl_xor(p, 4);
        p += __shfl_xor(p, 8);
        if (r == 0 && ov) atomicAdd(&l2t[q * NS + so], p);
    }
}

// ---------------------------------------------------------------------------
// Phase 3: logits[q,s] = -l2t[q,s]/500*scale ; out = log_softmax over s
// grid = 75, block = 32 (one wave per query row of 25)
// ---------------------------------------------------------------------------
__global__ void logsoftmax_kernel(const float* __restrict__ l2t,
                                  const float* __restrict__ sc,
                                  float* __restrict__ out) {
    const int q = blockIdx.x;
    const int t = threadIdx.x;
    const float scale = sc[0];

    float logit = (t < NS) ? (-l2t[q * NS + t] * LOGIT_SCALE * scale)
                           : -3.402823466e38f;

    float mx = logit;
#pragma unroll
    for (int mask = 1; mask < 32; mask <<= 1)
        mx = fmaxf(mx, __shfl_xor(mx, mask));

    float e = (t < NS) ? __expf(logit - mx) : 0.f;
#pragma unroll
    for (int mask = 1; mask < 32; mask <<= 1)
        e += __shfl_xor(e, mask);

    if (t < NS) out[q * NS + t] = logit - mx - __logf(e);
}

// ---------------------------------------------------------------------------
extern "C" void kernel_launch(void* const* d_in, const int* in_sizes, int n_in,
                              void* d_out, int out_size, void* d_ws, size_t ws_size,
                              hipStream_t stream) {
    const float* fm = (const float*)d_in[0];   // feature_map (100,640,25)
    const float* w2 = (const float*)d_in[1];   // weight2     (25,75,25)
    const float* sc = (const float*)d_in[2];   // scale       (1,)
    float* out = (float*)d_out;                // (75,25)

    float* cen = (float*)d_ws;                 // centroid raw (25,80,640) = 5.12 MB
    float* l2t = cen + NS * NQP * D;           // l2^T (75,25)

    zero_l2_kernel<<<dim3((NQ * NS + 255) / 256), 256, 0, stream>>>(l2t);
    centroid_kernel<<<dim3(DT, 5, NS), 32, 0, stream>>>(fm, w2, cen);
    qr_l2_kernel<<<dim3(DT, 2, NQ), 32, 0, stream>>>(fm, w2, cen, l2t);
    logsoftmax_kernel<<<NQ, 32, 0, stream>>>(l2t, sc, out);
}